// ForwardKinematicsLayer_64793876628144
// MI455X (gfx1250) — compile-verified
//
#include <hip/hip_runtime.h>
#include <cstdint>

// Forward kinematics: B=131072 items, J=24 joints, parent-chain of 3x4 affine
// transforms. Memory-bound (~189 MB total traffic -> ~8us @ 23.3 TB/s).
// Strategy: async global->LDS staging (coalesced b128 reads + free layout
// transpose via per-lane LDS dest addresses), ASYNCcnt double buffering,
// fully-unrolled in-register chain math per lane, vectorized NT b128 stores.

#define FK_J 24
#define FK_NG 2            // item-groups of 32 per block
#define FK_CHUNK_JOINTS 8  // 3 chunks of 8 joints
// Per-item LDS region (dwords): 72 rot + 24 pos = 96, padded to 100 (keeps
// 16B alignment for async b128 writes; 2-way-max bank conflicts on reads).
#define FK_ITEM_DW 100

typedef float v4f __attribute__((ext_vector_type(4)));

__device__ __forceinline__ uint32_t lds_off_u32(const void* p) {
    // Flat shared-aperture addresses carry the LDS byte offset in the low 32
    // bits (ISA 10.2).
    return (uint32_t)(uintptr_t)p;
}

// Issue one stage: copy 8 joints' rot+pos for 32 consecutive items into the
// given LDS buffer. Lanes 0..17 carry the 18 b128s of an item's rotation
// slice (288B), lanes 18..23 carry the 6 b128s of its position slice (96B).
// One predicated async-b128 per item => 32 ASYNCcnt increments per stage.
__device__ __forceinline__ void issue_chunk(const float* __restrict__ rot,
                                            const float* __restrict__ pos,
                                            int itemBase, int j0,
                                            uint32_t ldsBufByte, int lane) {
    if (lane < 24) {
        const bool isRot = lane < 18;
        uint32_t laneLds = isRot ? (uint32_t)lane * 16u
                                 : 288u + (uint32_t)(lane - 18) * 16u;
        uint64_t gbase;
        uint32_t gstride;
        if (isRot) {
            gbase = (uint64_t)(uintptr_t)rot +
                    (uint64_t)itemBase * 864ull + (uint32_t)j0 * 36u +
                    (uint32_t)lane * 16u;
            gstride = 864u;  // item stride in rotations[]
        } else {
            gbase = (uint64_t)(uintptr_t)pos +
                    (uint64_t)itemBase * 288ull + (uint32_t)j0 * 12u +
                    (uint32_t)(lane - 18) * 16u;
            gstride = 288u;  // item stride in positions[]
        }
        uint32_t lds = ldsBufByte + laneLds;
#pragma unroll 4
        for (int m = 0; m < 32; ++m) {
            asm volatile("global_load_async_to_lds_b128 %0, %1, off"
                         :: "v"(lds), "v"(gbase)
                         : "memory");
            lds += (uint32_t)(FK_ITEM_DW * 4);
            gbase += gstride;
        }
    }
}

__global__ __launch_bounds__(32) void fk_chain_kernel(
        const float* __restrict__ rot,   // [B][24][3][3]
        const float* __restrict__ pos,   // [B][24][3]
        float* __restrict__ out) {       // [B][24][3]
    constexpr int PAR[FK_J] = {0, 0, 0, 0, 1, 2, 3, 4, 5, 6, 7, 8,
                               9, 9, 9, 12, 13, 14, 16, 17, 18, 19, 20, 21};

    __shared__ __align__(16) float sbuf[2][32][FK_ITEM_DW];

    const int lane = threadIdx.x;  // wave32: one wave per block
    const int blockBase = blockIdx.x * (32 * FK_NG);

    const uint32_t ldsBase0 = lds_off_u32(&sbuf[0][0][0]);
    const uint32_t ldsBase1 = lds_off_u32(&sbuf[1][0][0]);

    // Prefetch stage 0 (group 0, joints 0..7) into buffer 0.
    issue_chunk(rot, pos, blockBase, 0, ldsBase0, lane);

    float R[FK_J][9];
    float T[FK_J][3];

    for (int grp = 0; grp < FK_NG; ++grp) {
        const int itemBase = blockBase + grp * 32;
        const int item = itemBase + lane;
        float* __restrict__ o = out + (size_t)item * (FK_J * 3);

#pragma unroll
        for (int c = 0; c < 3; ++c) {
            const int s = grp * 3 + c;  // stage index; buffer = s & 1

            // Wait for this stage's async copy to land in LDS.
            asm volatile("s_wait_asynccnt 0x0" ::: "memory");

            // Kick off the next stage into the other buffer; its copy
            // overlaps this stage's compute.
            {
                const int nGrp = (c == 2) ? grp + 1 : grp;
                const int nC = (c == 2) ? 0 : c + 1;
                if (nGrp < FK_NG) {
                    const uint32_t nBuf = ((s + 1) & 1) ? ldsBase1 : ldsBase0;
                    issue_chunk(rot, pos, blockBase + nGrp * 32, nC * 8,
                                nBuf, lane);
                }
            }

            const float* __restrict__ my =
                (s & 1) ? &sbuf[1][lane][0] : &sbuf[0][lane][0];

            // Chunk-local output staging: 8 joints x 3 coords = 24 dwords,
            // stored as 6 x b128 (item region 288B-aligned, chunk at +c*96B).
            float ob[FK_CHUNK_JOINTS * 3];

#pragma unroll
            for (int jj = 0; jj < FK_CHUNK_JOINTS; ++jj) {
                const int j = c * FK_CHUNK_JOINTS + jj;

                float l[9], p[3];
#pragma unroll
                for (int k = 0; k < 9; ++k) l[k] = my[jj * 9 + k];
#pragma unroll
                for (int k = 0; k < 3; ++k) p[k] = my[72 + jj * 3 + k];

                if (j == 0) {
#pragma unroll
                    for (int k = 0; k < 9; ++k) R[0][k] = l[k];
#pragma unroll
                    for (int k = 0; k < 3; ++k) T[0][k] = p[k];
                } else {
                    const int pa = PAR[j];
#pragma unroll
                    for (int r = 0; r < 3; ++r) {
                        const float a0 = R[pa][r * 3 + 0];
                        const float a1 = R[pa][r * 3 + 1];
                        const float a2 = R[pa][r * 3 + 2];
#pragma unroll
                        for (int cc = 0; cc < 3; ++cc) {
                            R[j][r * 3 + cc] =
                                fmaf(a0, l[0 * 3 + cc],
                                fmaf(a1, l[1 * 3 + cc],
                                     a2 * l[2 * 3 + cc]));
                        }
                        T[j][r] = fmaf(a0, p[0],
                                  fmaf(a1, p[1],
                                  fmaf(a2, p[2], T[pa][r])));
                    }
                }

                ob[jj * 3 + 0] = T[j][0];
                ob[jj * 3 + 1] = T[j][1];
                ob[jj * 3 + 2] = T[j][2];
            }

            // 6 x 16B non-temporal stores (write-once output; keep L2 for
            // the streamed input).
            {
                v4f* __restrict__ ov = (v4f*)(o + c * (FK_CHUNK_JOINTS * 3));
#pragma unroll
                for (int k = 0; k < 6; ++k) {
                    v4f v = {ob[k * 4 + 0], ob[k * 4 + 1],
                             ob[k * 4 + 2], ob[k * 4 + 3]};
                    __builtin_nontemporal_store(v, ov + k);
                }
            }
        }
    }
}

extern "C" void kernel_launch(void* const* d_in, const int* in_sizes, int n_in,
                              void* d_out, int out_size, void* d_ws, size_t ws_size,
                              hipStream_t stream) {
    (void)n_in; (void)out_size; (void)d_ws; (void)ws_size;
    const float* rot = (const float*)d_in[0];  // B*24*9 floats
    const float* pos = (const float*)d_in[1];  // B*24*3 floats
    float* out = (float*)d_out;                // B*24*3 floats

    const int B = in_sizes[0] / (FK_J * 9);
    const int itemsPerBlock = 32 * FK_NG;
    const int blocks = B / itemsPerBlock;  // B=131072 -> 2048 blocks

    fk_chain_kernel<<<blocks, 32, 0, stream>>>(rot, pos, out);
}